// cross_entropy_loss_28982439313956
// MI455X (gfx1250) — compile-verified
//
#include <hip/hip_runtime.h>
#include <hip/hip_bf16.h>
#include <cstdint>

// Problem constants (from reference): B=64, S=48, MMAX=24 -> K=25, V=2048
#define CE_B 64
#define CE_S 48
#define CE_MMAX 24
#define CE_K 25
#define CE_V 2048
#define CE_ROWS (CE_B * CE_S * CE_K)        // 76800
#define CE_WAVES_PER_BLOCK 4
#define CE_THREADS 128
#define CE_NBLOCKS (CE_ROWS / CE_WAVES_PER_BLOCK)  // 19200

// Stage 1: one wave32 per (b,s,k) row. Invalid rows never touch the logits.
// Valid rows: async-DMA the 8KB row HBM -> LDS (global_load_async_to_lds_b128,
// ASYNCcnt), wait, then one pass from LDS: row max + sum(exp) via wave32
// shuffles; label logit via uniform ds_load broadcast.
__global__ void __launch_bounds__(CE_THREADS)
ce_row_kernel(const int* __restrict__ labels,
              const float* __restrict__ logits,
              const int* __restrict__ seq_length,
              const int* __restrict__ mlen,
              const int* __restrict__ end_tok_p,
              float* __restrict__ partials)
{
    __shared__ float smem[CE_WAVES_PER_BLOCK][CE_V];  // 32 KB: one row per wave
    __shared__ float wsum[CE_WAVES_PER_BLOCK];

    const int w    = threadIdx.x >> 5;
    const int lane = threadIdx.x & 31;
    const int row  = blockIdx.x * CE_WAVES_PER_BLOCK + w;

    const int b   = row / (CE_S * CE_K);
    const int rem = row % (CE_S * CE_K);
    const int si  = rem / CE_K;
    const int k   = rem % CE_K;

    const int  m     = mlen[b * CE_S + si];
    const bool valid = (si < seq_length[b]) && (k <= m);  // uniform per wave

    float waveval = 0.0f;
    if (valid) {
        // ---- async copy: 16 x (32 lanes x 16B) = 8192 B, HBM -> LDS ----
        const float* gsrc  = logits + (size_t)row * CE_V;
        uint64_t gaddr = (uint64_t)(uintptr_t)gsrc + (uint64_t)(lane * 16);
        uint32_t laddr = (uint32_t)(uintptr_t)(&smem[w][0]) + (uint32_t)(lane * 16);
        asm volatile(
            "global_load_async_to_lds_b128 %0, %1, off\n\t"
            "global_load_async_to_lds_b128 %0, %1, off offset:512\n\t"
            "global_load_async_to_lds_b128 %0, %1, off offset:1024\n\t"
            "global_load_async_to_lds_b128 %0, %1, off offset:1536\n\t"
            "global_load_async_to_lds_b128 %0, %1, off offset:2048\n\t"
            "global_load_async_to_lds_b128 %0, %1, off offset:2560\n\t"
            "global_load_async_to_lds_b128 %0, %1, off offset:3072\n\t"
            "global_load_async_to_lds_b128 %0, %1, off offset:3584\n\t"
            "global_load_async_to_lds_b128 %0, %1, off offset:4096\n\t"
            "global_load_async_to_lds_b128 %0, %1, off offset:4608\n\t"
            "global_load_async_to_lds_b128 %0, %1, off offset:5120\n\t"
            "global_load_async_to_lds_b128 %0, %1, off offset:5632\n\t"
            "global_load_async_to_lds_b128 %0, %1, off offset:6144\n\t"
            "global_load_async_to_lds_b128 %0, %1, off offset:6656\n\t"
            "global_load_async_to_lds_b128 %0, %1, off offset:7168\n\t"
            "global_load_async_to_lds_b128 %0, %1, off offset:7680\n\t"
            :: "v"(laddr), "v"(gaddr) : "memory");
        asm volatile("s_wait_asynccnt 0x0" ::: "memory");

        // ---- one pass: 16 interleaved float4 per lane (conflict-free) ----
        const float4* s4 = reinterpret_cast<const float4*>(&smem[w][0]);
        float4 r[16];
#pragma unroll
        for (int j = 0; j < 16; ++j) r[j] = s4[j * 32 + lane];

        float mx = -3.402823466e38f;
#pragma unroll
        for (int j = 0; j < 16; ++j)
            mx = fmaxf(mx, fmaxf(fmaxf(r[j].x, r[j].y), fmaxf(r[j].z, r[j].w)));
#pragma unroll
        for (int off = 16; off > 0; off >>= 1)
            mx = fmaxf(mx, __shfl_xor(mx, off, 32));

        float se = 0.0f;
#pragma unroll
        for (int j = 0; j < 16; ++j) {
            se += __expf(r[j].x - mx);
            se += __expf(r[j].y - mx);
            se += __expf(r[j].z - mx);
            se += __expf(r[j].w - mx);
        }
#pragma unroll
        for (int off = 16; off > 0; off >>= 1)
            se += __shfl_xor(se, off, 32);

        const int endtok = end_tok_p[0];
        const int lab = (k == m) ? endtok : labels[(b * CE_S + si) * CE_MMAX + k];
        const float xl = smem[w][lab];   // uniform ds_load broadcast

        waveval = mx + __logf(se) - xl;  // nll = max + log(sum exp) - x_lab
    }

    if (lane == 0) wsum[w] = waveval;
    __syncthreads();
    if (threadIdx.x == 0)
        partials[blockIdx.x] = (wsum[0] + wsum[1]) + (wsum[2] + wsum[3]);
}

// Stage 2: single block; deterministic fixed-order reduction of partials,
// count computed analytically from seq_length / m_length_matrix.
__global__ void __launch_bounds__(256)
ce_finalize_kernel(const float* __restrict__ partials, int nparts,
                   const int* __restrict__ seq_length,
                   const int* __restrict__ mlen,
                   float* __restrict__ out)
{
    float s = 0.0f;
    for (int i = threadIdx.x; i < nparts; i += 256) s += partials[i];

    float c = 0.0f;
    for (int i = threadIdx.x; i < CE_B * CE_S; i += 256) {
        const int b  = i / CE_S;
        const int si = i % CE_S;
        if (si < seq_length[b]) c += (float)(mlen[i] + 1);
    }

#pragma unroll
    for (int off = 16; off > 0; off >>= 1) {
        s += __shfl_xor(s, off, 32);
        c += __shfl_xor(c, off, 32);
    }

    __shared__ float shs[8];
    __shared__ float shc[8];
    const int w    = threadIdx.x >> 5;
    const int lane = threadIdx.x & 31;
    if (lane == 0) { shs[w] = s; shc[w] = c; }
    __syncthreads();
    if (threadIdx.x == 0) {
        float S_ = 0.0f, C_ = 0.0f;
        for (int i = 0; i < 8; ++i) { S_ += shs[i]; C_ += shc[i]; }
        out[0] = S_ / C_;
    }
}

extern "C" void kernel_launch(void* const* d_in, const int* in_sizes, int n_in,
                              void* d_out, int out_size, void* d_ws, size_t ws_size,
                              hipStream_t stream)
{
    const int*   labels  = (const int*)d_in[0];     // (B,S,MMAX) int32
    const float* logits  = (const float*)d_in[1];   // (B,S,K,V) float32
    const int*   seq_len = (const int*)d_in[2];     // (B,) int32
    const int*   mlen    = (const int*)d_in[3];     // (B,S) int32
    const int*   end_tok = (const int*)d_in[5];     // scalar int32 (V-1)
    float*       out     = (float*)d_out;
    float*       parts   = (float*)d_ws;            // CE_NBLOCKS floats

    ce_row_kernel<<<CE_NBLOCKS, CE_THREADS, 0, stream>>>(
        labels, logits, seq_len, mlen, end_tok, parts);
    ce_finalize_kernel<<<1, 256, 0, stream>>>(
        parts, CE_NBLOCKS, seq_len, mlen, out);
}